// rlf_53446573032131
// MI455X (gfx1250) — compile-verified
//
#include <hip/hip_runtime.h>
#include <hip/hip_bf16.h>

// CDNA5 / gfx1250, wave32. All matmuls via V_WMMA_F32_16X16X4_F32.
typedef __attribute__((ext_vector_type(2))) float v2f;
typedef __attribute__((ext_vector_type(8))) float v8f;

#define NW 8          // waves per block (256 threads)
#define LEAKY 0.2f

// ---- repacked-weight layout in d_ws (floats) ----
// r = kz*(in+1)+ix  (kz in 0..32, ix in 0..in); row 32 / col `in` carry the bias terms.
#define OFF_WE1 0            // 100  x 64
#define OFF_WE2 6400         // 2148 x 64
#define OFF_WE3 143872       // 2148 x 16
#define OFF_WF1 178240       // 1092 x 128
#define OFF_WF2 318016       // 4260 x 128
#define OFF_WF3 863296       // 4260 x 16 (8 real cols, padded)

// -------- WMMA fragment helpers (f32 16x16x4, layouts per CDNA5 ISA 7.12.2) --------
// A (16x4): lanes 0-15 hold M=lane,K={k,k+1}; lanes 16-31 hold M=lane-16,K={k+2,k+3}
// B (4x16): VGPR0 holds K=k (lanes 0-15) / K=k+2 (lanes 16-31); VGPR1 = +1
// C/D     : VGPR v: lanes 0-15 -> M=v, lanes 16-31 -> M=v+8; N = lane&15

__device__ inline v2f afrag_rowptr(const float* rowp, int k, int lane) {
  const float* p = rowp + k + ((lane >> 4) << 1);
  v2f r; r.x = p[0]; r.y = p[1]; return r;
}

template<int IN1>  // IN1 = in+1 (compile-time so div/mod fold to mul/shift)
__device__ inline v2f afrag_outer(const float* zrow, const float* xrow, int k, int lane) {
  int kk = k + ((lane >> 4) << 1);
  v2f r;
#pragma unroll
  for (int j = 0; j < 2; ++j) {
    int rr = kk + j;
    int kz = rr / IN1;
    int ix = rr - kz * IN1;
    float zv = (kz < 32) ? zrow[kz] : ((kz == 32) ? 1.f : 0.f);
    float xv = (ix == IN1 - 1) ? 1.f : xrow[ix];
    r[j] = zv * xv;
  }
  return r;
}

__device__ inline v2f bfrag(const float* B, int ldn, int k, int n0, int lane) {
  const float* p = B + (size_t)(k + ((lane >> 4) << 1)) * ldn + n0 + (lane & 15);
  v2f r; r.x = p[0]; r.y = p[ldn]; return r;
}

// One GEMM stage: out(16 x ntiles*16) += A(16 x Rpad) * B(Rpad x ...), split-K into
// `kchunks`, work items (tile,kchunk) distributed over waves [wave_lo, wave_lo+wave_n),
// results accumulated into LDS with ds_add_f32 (outbuf pre-initialized with bias).
template<class AG>
__device__ inline void gemm_acc(AG&& ag, const float* B, int b_ld,
                                float* outbuf, int out_ld, int out_col0,
                                int Rpad, int ntiles, int kchunks,
                                int wave_lo, int wave_n, int wave, int lane)
{
  if (wave < wave_lo || wave >= wave_lo + wave_n) return;
  int steps_total = Rpad >> 2;
  int steps = (steps_total + kchunks - 1) / kchunks;
  int items = ntiles * kchunks;
  for (int it = wave - wave_lo; it < items; it += wave_n) {
    int t  = it % ntiles;
    int kc = it / ntiles;
    int k0 = kc * steps * 4;
    int k1 = k0 + steps * 4; if (k1 > Rpad) k1 = Rpad;
    v8f acc = {};
    for (int k = k0; k < k1; k += 4) {
      v2f a = ag(k);
      v2f b = bfrag(B, b_ld, k, t * 16, lane);
      acc = __builtin_amdgcn_wmma_f32_16x16x4_f32(false, a, false, b,
                                                  (short)0, acc, false, false);
    }
    int n   = out_col0 + t * 16 + (lane & 15);
    int mhi = (lane >> 4) << 3;
#pragma unroll
    for (int v = 0; v < 8; ++v)
      atomicAdd(&outbuf[(size_t)(v + mhi) * out_ld + n], acc[v]);
  }
}

__device__ inline void fill_bias(float* buf, const float* bias, int nout, int tid) {
  for (int i = tid; i < 16 * nout; i += 256) buf[i] = bias ? bias[i % nout] : 0.f;
}
__device__ inline void leaky_ip(float* buf, int n, int tid) {
  for (int i = tid; i < n; i += 256) { float v = buf[i]; buf[i] = v >= 0.f ? v : LEAKY * v; }
}

// -------- repack e_w/f_w into outer-product GEMM weight matrices --------
__global__ void rlf_repack(const float* __restrict__ w, const float* __restrict__ bvec,
                           float* __restrict__ dst, int D, int in, int out, int out_pad,
                           int off_w, int off_b, int Rpad)
{
  int i = blockIdx.x * blockDim.x + threadIdx.x;
  if (i >= Rpad * out_pad) return;
  int r = i / out_pad, o = i - r * out_pad;
  int in1 = in + 1;
  float val = 0.f;
  if (r < 33 * in1 && o < out) {
    int kz = r / in1;
    int ix = r - kz * in1;
    int col = (ix < in) ? (off_w + ix * out + o) : (off_b + o);
    val = (kz < 32) ? w[(size_t)kz * D + col] : bvec[col];
  }
  dst[i] = val;
}

// -------- fused per-16-batch pipeline --------
__global__ void __launch_bounds__(256)
rlf_fused(const int* __restrict__ indices, const float* __restrict__ states,
          const float* __restrict__ maps,
          const float* __restrict__ fc1_w, const float* __restrict__ fc1_b,
          const float* __restrict__ fc2_w, const float* __restrict__ fc2_b,
          const float* __restrict__ fc3_w, const float* __restrict__ fc3_b,
          const float* __restrict__ bn_w,  const float* __restrict__ bn_b,
          const float* __restrict__ ws, float* __restrict__ out)
{
  __shared__ float sm[8768];
  float* Z1  = sm;            // 16x256 (fc1 out)
  float* Z2  = sm + 4096;     // 16x128 (fc2 out)
  float* Z3  = sm + 6144;     // 16x128 (fc3 out)
  float* ZL  = sm + 8192;     // 16x32  (latent z, live to the end)
  float* ST  = sm + 8704;     // 16x4   (states tile)
  float* E1S = sm;            // reuse Z1 region (dead after fc2)
  float* E1G = sm + 1024;
  float* E2S = sm + 2048;
  float* E2G = sm + 3072;
  float* ZF  = sm + 4096;     // 16x32, reuse Z2 region (dead after fc3)
  float* F1  = sm + 6144;     // 16x128, reuse Z3 region (dead after bn)
  float* F2  = sm;            // 16x128, reuse E1 region (dead after e2)
  float* ANG = sm + 4096;     // 16x16, reuse ZF region (dead after f1)

  const int tid = threadIdx.x, wave = tid >> 5, lane = tid & 31, mrow = lane & 15;
  const int b0 = blockIdx.x * 16;
  const float* maprow = maps + (size_t)indices[b0 + mrow] * 1024;
  const float* zrow = ZL + mrow * 32;

  if (tid < 64) ST[tid] = states[(size_t)b0 * 4 + tid];

  // fc1: (16x1024)@(1024x256)+b, leaky
  fill_bias(Z1, fc1_b, 256, tid); __syncthreads();
  gemm_acc([&](int k){ return afrag_rowptr(maprow, k, lane); },
           fc1_w, 256, Z1, 256, 0, 1024, 16, 1, 0, 8, wave, lane);
  __syncthreads(); leaky_ip(Z1, 4096, tid); __syncthreads();

  // fc2: 256->128, leaky
  fill_bias(Z2, fc2_b, 128, tid); __syncthreads();
  gemm_acc([&](int k){ return afrag_rowptr(Z1 + mrow * 256, k, lane); },
           fc2_w, 128, Z2, 128, 0, 256, 8, 1, 0, 8, wave, lane);
  __syncthreads(); leaky_ip(Z2, 2048, tid); __syncthreads();

  // fc3: 128->128, leaky
  fill_bias(Z3, fc3_b, 128, tid); __syncthreads();
  gemm_acc([&](int k){ return afrag_rowptr(Z2 + mrow * 128, k, lane); },
           fc3_w, 128, Z3, 128, 0, 128, 8, 1, 0, 8, wave, lane);
  __syncthreads(); leaky_ip(Z3, 2048, tid); __syncthreads();

  // bn: 128->32 (no act); split-K x4 over 2 tiles
  fill_bias(ZL, bn_b, 32, tid); __syncthreads();
  gemm_acc([&](int k){ return afrag_rowptr(Z3 + mrow * 128, k, lane); },
           bn_w, 32, ZL, 32, 0, 128, 2, 4, 0, 8, wave, lane);
  __syncthreads();

  // e-layer1: in=2 (S|G), out=64; A = outer([z,1],[x,1]); bias folded into B
  fill_bias(E1S, nullptr, 64, tid); fill_bias(E1G, nullptr, 64, tid); __syncthreads();
  gemm_acc([&](int k){ return afrag_outer<3>(zrow, ST + mrow * 4 + 0, k, lane); },
           ws + OFF_WE1, 64, E1S, 64, 0, 100, 4, 1, 0, 4, wave, lane);
  gemm_acc([&](int k){ return afrag_outer<3>(zrow, ST + mrow * 4 + 2, k, lane); },
           ws + OFF_WE1, 64, E1G, 64, 0, 100, 4, 1, 4, 4, wave, lane);
  __syncthreads(); leaky_ip(E1S, 1024, tid); leaky_ip(E1G, 1024, tid); __syncthreads();

  // e-layer2: 64->64
  fill_bias(E2S, nullptr, 64, tid); fill_bias(E2G, nullptr, 64, tid); __syncthreads();
  gemm_acc([&](int k){ return afrag_outer<65>(zrow, E1S + mrow * 64, k, lane); },
           ws + OFF_WE2, 64, E2S, 64, 0, 2148, 4, 1, 0, 4, wave, lane);
  gemm_acc([&](int k){ return afrag_outer<65>(zrow, E1G + mrow * 64, k, lane); },
           ws + OFF_WE2, 64, E2G, 64, 0, 2148, 4, 1, 4, 4, wave, lane);
  __syncthreads(); leaky_ip(E2S, 1024, tid); leaky_ip(E2G, 1024, tid); __syncthreads();

  // e-layer3: 64->16 (no act), results concatenated into ZF[16x32]; split-K x4
  fill_bias(ZF, nullptr, 32, tid); __syncthreads();
  gemm_acc([&](int k){ return afrag_outer<65>(zrow, E2S + mrow * 64, k, lane); },
           ws + OFF_WE3, 16, ZF, 32, 0, 2148, 1, 4, 0, 4, wave, lane);
  gemm_acc([&](int k){ return afrag_outer<65>(zrow, E2G + mrow * 64, k, lane); },
           ws + OFF_WE3, 16, ZF, 32, 16, 2148, 1, 4, 4, 4, wave, lane);
  __syncthreads();

  // f-layer1: 32->128, leaky
  fill_bias(F1, nullptr, 128, tid); __syncthreads();
  gemm_acc([&](int k){ return afrag_outer<33>(zrow, ZF + mrow * 32, k, lane); },
           ws + OFF_WF1, 128, F1, 128, 0, 1092, 8, 1, 0, 8, wave, lane);
  __syncthreads(); leaky_ip(F1, 2048, tid); __syncthreads();

  // f-layer2: 128->128, leaky
  fill_bias(F2, nullptr, 128, tid); __syncthreads();
  gemm_acc([&](int k){ return afrag_outer<129>(zrow, F1 + mrow * 128, k, lane); },
           ws + OFF_WF2, 128, F2, 128, 0, 4260, 8, 1, 0, 8, wave, lane);
  __syncthreads(); leaky_ip(F2, 2048, tid); __syncthreads();

  // f-layer3: 128->8 (padded to 16 cols), split-K x8
  fill_bias(ANG, nullptr, 16, tid); __syncthreads();
  gemm_acc([&](int k){ return afrag_outer<129>(zrow, F2 + mrow * 128, k, lane); },
           ws + OFF_WF3, 16, ANG, 16, 0, 4260, 1, 8, 0, 8, wave, lane);
  __syncthreads();

  // pts = [sin(ang), cos(ang)] -> out[b, 0:16]
  {
    int m = tid >> 4, c = tid & 15;
    float a = ANG[m * 16 + (c < 8 ? c : c - 8)];
    out[(size_t)(b0 + m) * 16 + c] = (c < 8) ? sinf(a) : cosf(a);
  }
}

extern "C" void kernel_launch(void* const* d_in, const int* in_sizes, int n_in,
                              void* d_out, int out_size, void* d_ws, size_t ws_size,
                              hipStream_t stream) {
  const int*   indices = (const int*)  d_in[0];
  const float* states  = (const float*)d_in[1];
  const float* maps    = (const float*)d_in[2];
  const float* fc1_w   = (const float*)d_in[3];
  const float* fc1_b   = (const float*)d_in[4];
  const float* fc2_w   = (const float*)d_in[5];
  const float* fc2_b   = (const float*)d_in[6];
  const float* fc3_w   = (const float*)d_in[7];
  const float* fc3_b   = (const float*)d_in[8];
  const float* bn_w    = (const float*)d_in[9];
  const float* bn_b    = (const float*)d_in[10];
  const float* e_w     = (const float*)d_in[11];
  const float* e_b     = (const float*)d_in[12];
  const float* f_w     = (const float*)d_in[13];
  const float* f_b     = (const float*)d_in[14];
  float* ws = (float*)d_ws;

  auto rp = [&](const float* w, const float* b, float* dst, int D, int in, int out,
                int outp, int offw, int offb, int Rpad) {
    int total = Rpad * outp;
    rlf_repack<<<(total + 255) / 256, 256, 0, stream>>>(w, b, dst, D, in, out, outp,
                                                        offw, offb, Rpad);
  };
  rp(e_w, e_b, ws + OFF_WE1, 5456,   2,  64,  64,     0,   128,  100);
  rp(e_w, e_b, ws + OFF_WE2, 5456,  64,  64,  64,   192,  4288, 2148);
  rp(e_w, e_b, ws + OFF_WE3, 5456,  64,  16,  16,  4352,  5376, 2148);
  rp(f_w, f_b, ws + OFF_WF1, 21768, 32, 128, 128,     0,  4096, 1092);
  rp(f_w, f_b, ws + OFF_WF2, 21768, 128, 128, 128, 4224, 20608, 4260);
  rp(f_w, f_b, ws + OFF_WF3, 21768, 128,   8,  16, 20736, 21760, 4260);

  rlf_fused<<<512, 256, 0, stream>>>(indices, states, maps, fc1_w, fc1_b, fc2_w, fc2_b,
                                     fc3_w, fc3_b, bn_w, bn_b, ws, (float*)d_out);
}